// Net_10574209483064
// MI455X (gfx1250) — compile-verified
//
#include <hip/hip_runtime.h>
#include <hip/hip_bf16.h>

#define FEAT 32
#define TILE 256            // edges staged per block per step (== blockDim)
#define SCATTER_BLOCKS 1024

#if defined(__has_builtin)
#if __has_builtin(__builtin_amdgcn_global_load_async_to_lds_b32) && \
    __has_builtin(__builtin_amdgcn_s_wait_asynccnt)
#define USE_ASYNC_LDS 1
#endif
#endif

typedef __attribute__((address_space(1))) int g_i32;   // global
typedef __attribute__((address_space(3))) int l_i32;   // LDS

#define AS1I(p) ((g_i32*)(p))
#define AS3I(p) ((l_i32*)(p))

__device__ __forceinline__ float clampf(float v, float lo, float hi) {
    return fminf(fmaxf(v, lo), hi);
}

// lane broadcasts that lower to v_readlane_b32 (lane index must be literal)
__device__ __forceinline__ int bcast_i(int v, int l) {
    return __builtin_amdgcn_readlane(v, l);
}
__device__ __forceinline__ float bcast_f(float v, int l) {
    return __int_as_float(__builtin_amdgcn_readlane(__float_as_int(v), l));
}

// ---------------------------------------------------------------------------
// Precompute: masked src/dst indices and fused edge weight
//   w = (edge_weight*(1-pert)+pert)*edge_scale ; src/dst *= mask
// ---------------------------------------------------------------------------
__global__ void pre_kernel(const int* __restrict__ ei, const int* __restrict__ em,
                           const float* __restrict__ es, const float* __restrict__ pm,
                           const float* __restrict__ ew,
                           int* __restrict__ srcm, int* __restrict__ dstm,
                           float* __restrict__ w, int E) {
    int e = blockIdx.x * blockDim.x + threadIdx.x;
    if (e < E) {
        int m = em[e];
        srcm[e] = ei[e] * m;
        dstm[e] = ei[E + e] * m;
        float p = pm[e];
        w[e] = (ew[e] * (1.0f - p) + p) * es[e];
    }
}

__global__ void zero_kernel(float4* __restrict__ acc, int n4) {
    int i = blockIdx.x * blockDim.x + threadIdx.x;
    if (i < n4) acc[i] = make_float4(0.f, 0.f, 0.f, 0.f);
}

// ---------------------------------------------------------------------------
// Scatter: acc[dst*32+lane] += w * x[src*32+lane]
// Edge stream (src,dst,w) double-buffered through LDS with async loads.
// One wave32 = one edge at a time, lane == feature -> 128B coalesced ops.
// dst==0 (masked) edges -> per-wave register accumulation, single atomic
// flush at the end, avoiding the node-0 L2 atomic hotspot.
// ---------------------------------------------------------------------------
__global__ void __launch_bounds__(TILE)
scatter_kernel(const int* __restrict__ src, const int* __restrict__ dst,
               const float* __restrict__ w, const float* __restrict__ xin,
               float* __restrict__ acc, int E, int nTiles) {
    __shared__ int   ls[2][TILE];
    __shared__ int   ld[2][TILE];
    __shared__ float lw[2][TILE];

    const int t    = threadIdx.x;
    const int lane = t & 31;

    int tile = blockIdx.x;
    int buf  = 0;
    float sum0 = 0.0f;   // accumulation destined for node 0 (this lane's feature)

    // stage first tile
    if (tile < nTiles) {
        int idx = tile * TILE + t;
        if (idx >= E) idx = E - 1;
#if defined(USE_ASYNC_LDS)
        __builtin_amdgcn_global_load_async_to_lds_b32(AS1I(src + idx), AS3I(&ls[0][t]), 0, 0);
        __builtin_amdgcn_global_load_async_to_lds_b32(AS1I(dst + idx), AS3I(&ld[0][t]), 0, 0);
        __builtin_amdgcn_global_load_async_to_lds_b32(AS1I(w   + idx), AS3I(&lw[0][t]), 0, 0);
#else
        ls[0][t] = src[idx];
        ld[0][t] = dst[idx];
        lw[0][t] = w[idx];
#endif
    }

    for (; tile < nTiles; tile += gridDim.x) {
        int nextTile = tile + gridDim.x;
        if (nextTile < nTiles) {
            int idx = nextTile * TILE + t;
            if (idx >= E) idx = E - 1;
            int nb = buf ^ 1;
#if defined(USE_ASYNC_LDS)
            __builtin_amdgcn_global_load_async_to_lds_b32(AS1I(src + idx), AS3I(&ls[nb][t]), 0, 0);
            __builtin_amdgcn_global_load_async_to_lds_b32(AS1I(dst + idx), AS3I(&ld[nb][t]), 0, 0);
            __builtin_amdgcn_global_load_async_to_lds_b32(AS1I(w   + idx), AS3I(&lw[nb][t]), 0, 0);
            __builtin_amdgcn_s_wait_asynccnt(3);   // current tile's 3 loads done
#else
            ls[nb][t] = src[idx];
            ld[nb][t] = dst[idx];
            lw[nb][t] = w[idx];
#endif
        } else {
#if defined(USE_ASYNC_LDS)
            __builtin_amdgcn_s_wait_asynccnt(0);
#endif
        }
        __syncthreads();

        // consume: this lane owns edge (tileBase + t); wave iterates its 32 edges
        int   myS = ls[buf][t];
        int   myD = ld[buf][t];
        float myW = lw[buf][t];
        const int waveBase = tile * TILE + (t & ~31);
        if (waveBase + 32 <= E) {
            // full-tile fast path: full unroll -> literal-lane v_readlane,
            // scalar dst/branch, no per-edge bound check
            #pragma unroll
            for (int j = 0; j < 32; ++j) {
                int   s  = bcast_i(myS, j);
                int   d  = bcast_i(myD, j);
                float wv = bcast_f(myW, j);
                float v  = xin[s * FEAT + lane];
                if (d == 0) {                       // wave-uniform scalar branch
                    sum0 += wv * v;
                } else {
                    unsafeAtomicAdd(&acc[d * FEAT + lane], wv * v);
                }
            }
        } else {
            #pragma unroll 1
            for (int j = 0; j < 32; ++j) {
                if (waveBase + j >= E) break;
                int   s  = __shfl(myS, j, 32);
                int   d  = __shfl(myD, j, 32);
                float wv = __shfl(myW, j, 32);
                float v  = xin[s * FEAT + lane];
                if (d == 0) {
                    sum0 += wv * v;
                } else {
                    unsafeAtomicAdd(&acc[d * FEAT + lane], wv * v);
                }
            }
        }
        __syncthreads();
        buf ^= 1;
    }

    // flush node-0 contribution: one atomic per wave instead of one per edge
    unsafeAtomicAdd(&acc[lane], sum0);
}

// ---------------------------------------------------------------------------
// Combine: x' = clip(x + clip(acc - x, -1, 1), 0, 2); acc := 0 for next round
// ---------------------------------------------------------------------------
__global__ void combine_kernel(float4* __restrict__ acc, const float4* __restrict__ xin,
                               float4* __restrict__ xout, int n4) {
    int i = blockIdx.x * blockDim.x + threadIdx.x;
    if (i < n4) {
        float4 a = acc[i];
        float4 x = xin[i];
        float4 r;
        r.x = clampf(x.x + clampf(a.x - x.x, -1.f, 1.f), 0.f, 2.f);
        r.y = clampf(x.y + clampf(a.y - x.y, -1.f, 1.f), 0.f, 2.f);
        r.z = clampf(x.z + clampf(a.z - x.z, -1.f, 1.f), 0.f, 2.f);
        r.w = clampf(x.w + clampf(a.w - x.w, -1.f, 1.f), 0.f, 2.f);
        xout[i] = r;
        acc[i] = make_float4(0.f, 0.f, 0.f, 0.f);
    }
}

extern "C" void kernel_launch(void* const* d_in, const int* in_sizes, int n_in,
                              void* d_out, int out_size, void* d_ws, size_t ws_size,
                              hipStream_t stream) {
    const float* x  = (const float*)d_in[0];
    const int*   ei = (const int*)  d_in[1];
    const int*   em = (const int*)  d_in[2];
    const float* es = (const float*)d_in[3];
    const float* pm = (const float*)d_in[4];
    const float* ew = (const float*)d_in[5];
    float* out = (float*)d_out;

    const int E  = in_sizes[5];   // edge count
    const int NF = out_size;      // N * FEAT
    const int n4 = NF / 4;

    float* ws   = (float*)d_ws;
    float* w    = ws;                                  // E floats
    int*   srcm = (int*)(ws + (size_t)E);              // E ints
    int*   dstm = (int*)(ws + 2 * (size_t)E);          // E ints
    float* acc  = ws + 3 * (size_t)E;                  // NF floats
    float* xA   = acc + (size_t)NF;                    // NF floats
    float* xB   = xA  + (size_t)NF;                    // NF floats

    pre_kernel<<<(E + 255) / 256, 256, 0, stream>>>(ei, em, es, pm, ew, srcm, dstm, w, E);
    zero_kernel<<<(n4 + 255) / 256, 256, 0, stream>>>((float4*)acc, n4);

    const int nTiles = (E + TILE - 1) / TILE;
    const int sblocks = nTiles < SCATTER_BLOCKS ? nTiles : SCATTER_BLOCKS;

    const float* cur = x;
    for (int it = 0; it < 5; ++it) {
        scatter_kernel<<<sblocks, TILE, 0, stream>>>(srcm, dstm, w, cur, acc, E, nTiles);
        float* nxt = (it == 4) ? out : ((it & 1) ? xB : xA);
        combine_kernel<<<(n4 + 255) / 256, 256, 0, stream>>>(
            (float4*)acc, (const float4*)cur, (float4*)nxt, n4);
        cur = nxt;
    }
}